// Rz_layer_41369124995655
// MI455X (gfx1250) — compile-verified
//
#include <hip/hip_runtime.h>
#include <cmath>

#define N_QUBITS 14
#define DIM      16384
#define NB       16
#define BDIM     32
#define DC       2

typedef __attribute__((ext_vector_type(2))) float v2f;
typedef __attribute__((ext_vector_type(4))) float f4;
typedef __attribute__((ext_vector_type(8))) float v8f;

__device__ __forceinline__ float sign_of(int d, int q) {
  // s(d,q) = 1 - 2*bit_{13-q}(d); 0-pad for q >= 14 (K padded to 16)
  if (q >= N_QUBITS) return 0.0f;
  return 1.0f - 2.0f * (float)((d >> (N_QUBITS - 1 - q)) & 1);
}

// ---------------------------------------------------------------------------
// Kernel 1: phase table via V_WMMA_F32_16X16X4_F32.
//   C[m][n] = sum_q signs[d0+m][q] * w[n][q]   (16x16 tile, K=16 padded)
//   table[(n*DIM + d0+m)*2 + {0,1}] = {cos, sin} of (-0.5 * C[m][n])
// One wave (32 threads) per 16-row d-tile. EXEC all ones, wave32.
// ---------------------------------------------------------------------------
__global__ void __launch_bounds__(32) rz_phase_table_wmma(
    const float* __restrict__ w, float* __restrict__ table) {
  const int lane = threadIdx.x;
  const int half = lane >> 4;        // 0: lanes 0-15, 1: lanes 16-31
  const int mn   = lane & 15;        // A row m (for sourcing) / C column n (=k)
  const int dRow = blockIdx.x * 16 + mn;

  v8f acc = {0.f, 0.f, 0.f, 0.f, 0.f, 0.f, 0.f, 0.f};
#pragma unroll
  for (int s = 0; s < 4; ++s) {
    // 32-bit A 16x4 layout: VGPR0 holds K=0 (lanes 0-15) / K=2 (lanes 16-31),
    // VGPR1 holds K=1 / K=3.  Global q for this lane's a.x is 4s + 2*half.
    const int q0 = 4 * s + 2 * half;
    v2f a, b;
    a.x = sign_of(dRow, q0);
    a.y = sign_of(dRow, q0 + 1);
    // B (4x16) mirrors: VGPR0 = row K=0 / K=2 across lanes (N = lane&15).
    b.x = (q0     < N_QUBITS) ? w[mn * N_QUBITS + q0    ] : 0.0f;
    b.y = (q0 + 1 < N_QUBITS) ? w[mn * N_QUBITS + q0 + 1] : 0.0f;
    acc = __builtin_amdgcn_wmma_f32_16x16x4_f32(false, a, false, b,
                                                (short)0, acc, false, false);
  }

  // C/D layout: lane holds column n = lane&15; VGPR r holds row m = r + 8*half.
  const int k = mn;
#pragma unroll
  for (int r = 0; r < 8; ++r) {
    const int d = blockIdx.x * 16 + r + 8 * half;
    const float phase = -0.5f * acc[r];
    float sv, cv;
    sincosf(phase, &sv, &cv);
    float2* t = (float2*)(table + 2 * (k * DIM + d));
    *t = make_float2(cv, sv);
  }
}

// ---------------------------------------------------------------------------
// Kernel 2: HBM-bound streaming apply. 4 consecutive d per thread (b128 I/O),
// table (L2-resident) read once per thread, nontemporal on the 256MB stream.
// grid = B*NB*(DIM/1024) = 8192 blocks of 256 threads.
// ---------------------------------------------------------------------------
__global__ void __launch_bounds__(256) rz_apply_table(
    const float* __restrict__ sr, const float* __restrict__ si,
    const float* __restrict__ table, float* __restrict__ out) {
  const int blk   = blockIdx.x;
  const int chunk = blk & 15;
  const int k     = (blk >> 4) & 15;
  const int b     = blk >> 8;
  const int d0    = chunk * 1024 + threadIdx.x * 4;

  const f4* t4 = (const f4*)(table + 2 * (k * DIM + d0));
  const f4 t01 = t4[0];   // {cos0,sin0,cos1,sin1}
  const f4 t23 = t4[1];   // {cos2,sin2,cos3,sin3}

#pragma unroll
  for (int o = 0; o < DC; ++o) {
    const int base = ((b * NB + k) * DC + o) * DIM + d0;
    const f4 re = __builtin_nontemporal_load((const f4*)(sr + base));
    const f4 im = __builtin_nontemporal_load((const f4*)(si + base));
    f4 o0, o1;
    o0.x = re.x * t01.x - im.x * t01.y;  o0.y = re.x * t01.y + im.x * t01.x;
    o0.z = re.y * t01.z - im.y * t01.w;  o0.w = re.y * t01.w + im.y * t01.z;
    o1.x = re.z * t23.x - im.z * t23.y;  o1.y = re.z * t23.y + im.z * t23.x;
    o1.z = re.w * t23.z - im.w * t23.w;  o1.w = re.w * t23.w + im.w * t23.z;
    f4* op = (f4*)(out + 2 * (size_t)base);
    __builtin_nontemporal_store(o0, op);
    __builtin_nontemporal_store(o1, op + 1);
  }
}

// ---------------------------------------------------------------------------
// Fallback (only if ws_size < 2MB): compute phases inline, 4 sincos/thread.
// ---------------------------------------------------------------------------
__global__ void __launch_bounds__(256) rz_apply_inline(
    const float* __restrict__ sr, const float* __restrict__ si,
    const float* __restrict__ w, float* __restrict__ out) {
  const int blk   = blockIdx.x;
  const int chunk = blk & 15;
  const int k     = (blk >> 4) & 15;
  const int b     = blk >> 8;
  const int d0    = chunk * 1024 + threadIdx.x * 4;

  const float* wk = w + k * N_QUBITS;
  float s = 0.0f;
#pragma unroll
  for (int q = 0; q < 12; ++q)
    s += wk[q] * (1.0f - 2.0f * (float)((d0 >> (13 - q)) & 1));
  const float w12 = wk[12], w13 = wk[13];
  const float ph[4] = { s + w12 + w13, s + w12 - w13,
                        s - w12 + w13, s - w12 - w13 };
  float cv[4], sv[4];
#pragma unroll
  for (int j = 0; j < 4; ++j) sincosf(-0.5f * ph[j], &sv[j], &cv[j]);
  const f4 t01 = { cv[0], sv[0], cv[1], sv[1] };
  const f4 t23 = { cv[2], sv[2], cv[3], sv[3] };

#pragma unroll
  for (int o = 0; o < DC; ++o) {
    const int base = ((b * NB + k) * DC + o) * DIM + d0;
    const f4 re = __builtin_nontemporal_load((const f4*)(sr + base));
    const f4 im = __builtin_nontemporal_load((const f4*)(si + base));
    f4 o0, o1;
    o0.x = re.x * t01.x - im.x * t01.y;  o0.y = re.x * t01.y + im.x * t01.x;
    o0.z = re.y * t01.z - im.y * t01.w;  o0.w = re.y * t01.w + im.y * t01.z;
    o1.x = re.z * t23.x - im.z * t23.y;  o1.y = re.z * t23.y + im.z * t23.x;
    o1.z = re.w * t23.z - im.w * t23.w;  o1.w = re.w * t23.w + im.w * t23.z;
    f4* op = (f4*)(out + 2 * (size_t)base);
    __builtin_nontemporal_store(o0, op);
    __builtin_nontemporal_store(o1, op + 1);
  }
}

extern "C" void kernel_launch(void* const* d_in, const int* in_sizes, int n_in,
                              void* d_out, int out_size, void* d_ws, size_t ws_size,
                              hipStream_t stream) {
  (void)in_sizes; (void)n_in; (void)out_size;
  const float* sr = (const float*)d_in[0];   // [32,16,2,16384] f32
  const float* si = (const float*)d_in[1];   // [32,16,2,16384] f32
  const float* w  = (const float*)d_in[2];   // [16,14] f32
  float* out = (float*)d_out;                // complex64 interleaved

  const size_t tableBytes = (size_t)NB * DIM * 2 * sizeof(float);  // 2 MB
  const int mainGrid = BDIM * NB * (DIM / (256 * 4));              // 8192

  if (ws_size >= tableBytes) {
    float* table = (float*)d_ws;
    rz_phase_table_wmma<<<DIM / 16, 32, 0, stream>>>(w, table);
    rz_apply_table<<<mainGrid, 256, 0, stream>>>(sr, si, table, out);
  } else {
    rz_apply_inline<<<mainGrid, 256, 0, stream>>>(sr, si, w, out);
  }
}